// GNN_3882650436636
// MI455X (gfx1250) — compile-verified
//
#include <hip/hip_runtime.h>
#include <hip/hip_bf16.h>

typedef __attribute__((ext_vector_type(8)))  float  v8f;
typedef __attribute__((ext_vector_type(2)))  float  v2f;

#define D 128  // feature width (D_IN == D_HID == 128)

// ---------------------------------------------------------------------------
// Zero the aggregation buffer and the in-degree counters (grid-stride).
// ---------------------------------------------------------------------------
__global__ void sage_zero(float* __restrict__ agg, float* __restrict__ cnt,
                          long nAgg, long nCnt) {
  long i = (long)blockIdx.x * blockDim.x + threadIdx.x;
  long stride = (long)gridDim.x * blockDim.x;
  for (long j = i; j < nAgg; j += stride) agg[j] = 0.0f;
  for (long j = i; j < nCnt; j += stride) cnt[j] = 0.0f;
}

// ---------------------------------------------------------------------------
// Edge scatter: one wave (32 lanes) per edge. Each lane moves one float4
// (4 feats) of h[src] into agg[dst] via global_atomic_add_f32. Lane 0 bumps
// the degree counter. agg (51 MB) fits in the 192 MB L2, so the RMW traffic
// stays on-chip; the random 512 B row gathers are the HBM-bound part.
// ---------------------------------------------------------------------------
__global__ void sage_scatter(const float* __restrict__ h,
                             const int* __restrict__ src,
                             const int* __restrict__ dst,
                             float* __restrict__ agg,
                             float* __restrict__ cnt, int E) {
  int wid  = (int)((blockIdx.x * (long)blockDim.x + threadIdx.x) >> 5);
  int lane = threadIdx.x & 31;
  if (wid >= E) return;
  int s = src[wid];
  int d = dst[wid];
  const float4* xr = reinterpret_cast<const float4*>(h + (size_t)s * D);
  float* ar = agg + (size_t)d * D + lane * 4;
  __builtin_prefetch(ar, 1, 0);   // global_prefetch_b8 on the RMW target line
  float4 v = xr[lane];            // 32 lanes x float4 == 128 floats, coalesced
  atomicAdd(ar + 0, v.x);
  atomicAdd(ar + 1, v.y);
  atomicAdd(ar + 2, v.z);
  atomicAdd(ar + 3, v.w);
  if (lane == 0) atomicAdd(cnt + d, 1.0f);
}

// ---------------------------------------------------------------------------
// mean = agg / max(cnt, 1)
// ---------------------------------------------------------------------------
__global__ void sage_mean(float* __restrict__ agg, const float* __restrict__ cnt,
                          long total) {
  long i = (long)blockIdx.x * blockDim.x + threadIdx.x;
  long stride = (long)gridDim.x * blockDim.x;
  for (long j = i; j < total; j += stride) {
    float c = cnt[j >> 7];                 // j / 128
    agg[j] = agg[j] / fmaxf(c, 1.0f);
  }
}

// ---------------------------------------------------------------------------
// Fused SAGE linear: out = mean @ W_l + xin @ W_r + b  (+ optional ReLU)
// One block per 16-node M-tile; 8 waves, each owning one 16-wide N-tile.
// Logical A = [mean | xin] (16x256), logical W = [W_l ; W_r] (256x16/wave).
// Full fp32 via V_WMMA_F32_16X16X4_F32 (64 WMMAs / tile; GEMM is ~13 GFLOP
// total, negligible vs. the edge phase, so exact fp32 costs nothing).
// REQUIREMENT: grid supplies exact 16-row tiles (N % 16 == 0 here), so EXEC
// is all-1s around every WMMA and the epilogue needs no predication.
// ---------------------------------------------------------------------------
template <int RELU>
__global__ __launch_bounds__(256) void sage_gemm(
    const float* __restrict__ mean, const float* __restrict__ xin,
    const float* __restrict__ Wl, const float* __restrict__ Wr,
    const float* __restrict__ bias, float* __restrict__ out, int N) {
  __shared__ float sA[16][264];            // 16 rows x 256 K (padded), fp32

  const int Mbase = blockIdx.x * 16;
  const int tid = threadIdx.x;

  // Cooperatively stage the A tile: cols 0..127 = mean row, 128..255 = x row.
  for (int i = tid; i < 16 * 256; i += 256) {
    int r = i >> 8;
    int c = i & 255;
    int row = Mbase + r;
    row = (row < N) ? row : (N - 1);       // branchless clamp (never hits here)
    float v = (c < D) ? mean[(size_t)row * D + c]
                      : xin[(size_t)row * D + (c - D)];
    sA[r][c] = v;
  }
  __syncthreads();

  const int lane = tid & 31;
  const int wave = tid >> 5;               // N-tile index, 0..7
  const int col  = wave * 16 + (lane & 15);// global output column
  const int half = lane >> 4;              // 0: lanes 0-15, 1: lanes 16-31
  const int rowA = lane & 15;              // A-fragment row (M)

  v8f acc = {};                            // 16x16 fp32 accumulator (8 VGPRs)

  // K=4 per step; A lane layout: v0=K(2*half), v1=K(2*half+1); B mirrors.
  for (int k0 = 0; k0 < 256; k0 += 4) {
    const float* W = (k0 < D) ? Wl : Wr;
    const int kb = (k0 & (D - 1)) + half * 2;
    v2f a, b;
    a[0] = sA[rowA][k0 + half * 2];
    a[1] = sA[rowA][k0 + half * 2 + 1];
    b[0] = W[(size_t)kb * D + col];
    b[1] = W[(size_t)(kb + 1) * D + col];
    acc = __builtin_amdgcn_wmma_f32_16x16x4_f32(
        false, a, false, b, (short)0, acc, false, false);
  }

  // C layout: VGPR v -> row v + 8*half, col = lane&15 within the N-tile.
  const float bv = bias[col];
  float* orow = out + (size_t)(Mbase + half * 8) * D + col;
#pragma unroll
  for (int v = 0; v < 8; ++v) {
    float val = acc[v] + bv;
    if (RELU) val = fmaxf(val, 0.0f);
    orow[(size_t)v * D] = val;             // unpredicated: tiles are exact
  }
}

// ---------------------------------------------------------------------------
// Host-side orchestration (all on `stream`, graph-capture safe).
// ---------------------------------------------------------------------------
extern "C" void kernel_launch(void* const* d_in, const int* in_sizes, int n_in,
                              void* d_out, int out_size, void* d_ws, size_t ws_size,
                              hipStream_t stream) {
  const float* x   = (const float*)d_in[0];
  const int*   ei  = (const int*)d_in[1];
  const float* Wl1 = (const float*)d_in[2];
  const float* Wr1 = (const float*)d_in[3];
  const float* b1  = (const float*)d_in[4];
  const float* Wl2 = (const float*)d_in[5];
  const float* Wr2 = (const float*)d_in[6];
  const float* b2  = (const float*)d_in[7];
  float* out = (float*)d_out;

  const int N = in_sizes[0] / D;       // 100000
  const int E = in_sizes[1] / 2;       // 1600000
  const int* src = ei;
  const int* dst = ei + E;

  float* agg = (float*)d_ws;                    // N*128 f32
  float* cnt = agg + (size_t)N * D;             // N f32
  float* h   = cnt + N;                         // N*128 f32 (hidden layer)

  const long nAgg = (long)N * D;
  dim3 blk(256);
  const int zeroBlocks    = 4096;
  const int scatterBlocks = (E + 7) / 8;        // 8 waves (edges) per block
  const int meanBlocks    = 4096;
  const int gemmBlocks    = (N + 15) / 16;      // exact: 6250 (N % 16 == 0)

  // ---- Layer 1: h = relu(mean(x) @ Wl1 + b1 + x @ Wr1) ----
  sage_zero    <<<zeroBlocks,    blk, 0, stream>>>(agg, cnt, nAgg, N);
  sage_scatter <<<scatterBlocks, blk, 0, stream>>>(x, src, dst, agg, cnt, E);
  sage_mean    <<<meanBlocks,    blk, 0, stream>>>(agg, cnt, nAgg);
  sage_gemm<1> <<<gemmBlocks,    blk, 0, stream>>>(agg, x, Wl1, Wr1, b1, h, N);

  // ---- Layer 2: out = mean(h) @ Wl2 + b2 + h @ Wr2 ----
  sage_zero    <<<zeroBlocks,    blk, 0, stream>>>(agg, cnt, nAgg, N);
  sage_scatter <<<scatterBlocks, blk, 0, stream>>>(h, src, dst, agg, cnt, E);
  sage_mean    <<<meanBlocks,    blk, 0, stream>>>(agg, cnt, nAgg);
  sage_gemm<0> <<<gemmBlocks,    blk, 0, stream>>>(agg, h, Wl2, Wr2, b2, out, N);
}